// SelectionNet_37692632990275
// MI455X (gfx1250) — compile-verified
//
#include <hip/hip_runtime.h>

typedef __attribute__((ext_vector_type(2))) float v2f;
typedef __attribute__((ext_vector_type(8))) float v8f;

#define B_     32
#define T_     32
#define NP_    196
#define C_     768
#define D_     1536
#define S_     42
#define P_     10
#define RPB_   48           // padded rows per batch
#define ROWS_  1536         // 32 * 48
#define QKVN_  4608
#define HD_    512
#define NC_    150528       // N*C
#define KSEL_  16
#define NSAMP_ 500
#define AST    772          // LDS A-tile row stride (768 + 4 pad; 772%64==4 -> no bank conflicts)

// ---------------- K0: build prompt rows + zero pad rows of h ----------------
__global__ void k_init_h(float* __restrict__ h, const float* __restrict__ prompt) {
  int b = blockIdx.x, p = blockIdx.y;
  if (p < P_) {
    size_t row = (size_t)b * RPB_ + p;
    for (int c = threadIdx.x; c < D_; c += blockDim.x)
      h[row * D_ + c] = prompt[p * D_ + c];
  } else {
    size_t row = (size_t)b * RPB_ + S_ + (p - P_);
    for (int c = threadIdx.x; c < D_; c += blockDim.x)
      h[row * D_ + c] = 0.f;
  }
}

// ---------------- K1: mean/max pool over patches -> feat rows of h ----------
__global__ void k_pool(const float* __restrict__ x, float* __restrict__ h) {
  int bt = blockIdx.x;
  int b = bt >> 5, t = bt & 31;
  const float* xp = x + ((size_t)b * T_ * NP_ + (size_t)t * NP_) * C_;
  size_t row = (size_t)b * RPB_ + P_ + t;
  for (int c = threadIdx.x; c < C_; c += blockDim.x) {
    float s = 0.f, mx = -3.402823466e38f;
    const float* p = xp + c;
    for (int n = 0; n < NP_; ++n) {
      if ((n & 15) == 0) __builtin_prefetch(p + (size_t)(n + 16) * C_, 0, 0);
      float v = p[(size_t)n * C_];
      s += v;
      mx = fmaxf(mx, v);
    }
    h[row * D_ + c]      = s * (1.0f / NP_);
    h[row * D_ + C_ + c] = mx;
  }
}

// ---------------- LayerNorm over rows ---------------------------------------
__global__ void k_ln(const float* __restrict__ h, const float* __restrict__ g,
                     const float* __restrict__ bb, float* __restrict__ z) {
  __shared__ float red[256];
  int r = blockIdx.x, tid = threadIdx.x;
  const float* hr = h + (size_t)r * D_;
  float s = 0.f, q = 0.f;
  for (int c = tid; c < D_; c += 256) { float v = hr[c]; s += v; q += v * v; }
  red[tid] = s; __syncthreads();
  for (int o = 128; o > 0; o >>= 1) { if (tid < o) red[tid] += red[tid + o]; __syncthreads(); }
  float mean = red[0] * (1.f / D_);
  __syncthreads();
  red[tid] = q; __syncthreads();
  for (int o = 128; o > 0; o >>= 1) { if (tid < o) red[tid] += red[tid + o]; __syncthreads(); }
  float var = red[0] * (1.f / D_) - mean * mean;
  float rstd = rsqrtf(var + 1e-5f);
  float* zr = z + (size_t)r * D_;
  for (int c = tid; c < D_; c += 256)
    zr[c] = (hr[c] - mean) * rstd * g[c] + bb[c];
}

// ---------------- WMMA f32 GEMM: D = A[1536xD_] @ Bm[D_ x Ncols] + bias (+Cin)
// grid = (96 m-tiles, Ncols/128), 256 threads = 8 waves, one 16x16 tile/wave.
__global__ void k_gemm(const float* __restrict__ A, const float* __restrict__ Bm,
                       const float* __restrict__ bias, const float* __restrict__ Cin,
                       float* __restrict__ Dm, int Ncols) {
  __shared__ float As[16 * AST];
  int tid = threadIdx.x;
  int wid = tid >> 5, lane = tid & 31;
  int rowBase = blockIdx.x * 16;
  int colBase = (blockIdx.y * 8 + wid) * 16;
  int m = lane & 15, half = lane >> 4, koff = half * 2;
  int col = colBase + m;

  v8f acc;
#pragma unroll
  for (int r = 0; r < 8; ++r) {
    size_t row = (size_t)rowBase + r + half * 8;
    float iv = bias ? bias[col] : 0.f;
    if (Cin) iv += Cin[row * Ncols + col];
    acc[r] = iv;
  }

  for (int ph = 0; ph < 2; ++ph) {
    int k0 = ph * 768;
    __syncthreads();
    for (int r = 0; r < 16; ++r)
      for (int c = tid; c < 768; c += 256)
        As[r * AST + c] = A[((size_t)rowBase + r) * D_ + k0 + c];
    __syncthreads();
    for (int kb = 0; kb < 768; kb += 4) {
      v2f av, bv;
      av.x = As[m * AST + kb + koff];
      av.y = As[m * AST + kb + koff + 1];
      const float* bp = Bm + ((size_t)k0 + kb + koff) * Ncols + col;
      bv.x = bp[0];
      bv.y = bp[Ncols];
      acc = __builtin_amdgcn_wmma_f32_16x16x4_f32(false, av, false, bv,
                                                  (short)0, acc, false, false);
    }
  }
#pragma unroll
  for (int r = 0; r < 8; ++r)
    Dm[((size_t)rowBase + r + half * 8) * Ncols + col] = acc[r];
}

// ---------------- attention per (batch, head) -------------------------------
__global__ void k_attn(const float* __restrict__ qkv, float* __restrict__ o) {
  __shared__ float att[S_ * 44];
  int b = blockIdx.x, head = blockIdx.y, tid = threadIdx.x;
  const float scale = 0.04419417382415922f; // 512^-0.5
  const float* base = qkv + (size_t)b * RPB_ * QKVN_ + head * HD_;

  for (int p = tid; p < S_ * S_; p += 256) {
    int i = p / S_, j = p % S_;
    const float* qi = base + (size_t)i * QKVN_;
    const float* kj = base + (size_t)j * QKVN_ + D_;
    float s = 0.f;
    for (int d = 0; d < HD_; ++d) s += qi[d] * kj[d];
    att[i * 44 + j] = s * scale;
  }
  __syncthreads();
  for (int i = tid; i < S_; i += 256) {
    float mx = -3.402823466e38f;
    for (int j = 0; j < S_; ++j) mx = fmaxf(mx, att[i * 44 + j]);
    float sum = 0.f;
    for (int j = 0; j < S_; ++j) { float e = __expf(att[i * 44 + j] - mx); att[i * 44 + j] = e; sum += e; }
    float inv = 1.f / sum;
    for (int j = 0; j < S_; ++j) att[i * 44 + j] *= inv;
  }
  __syncthreads();
  for (int idx = tid; idx < S_ * HD_; idx += 256) {
    int i = idx >> 9, d = idx & 511;
    const float* vb = base + 2 * D_ + d;
    float s = 0.f;
    for (int j = 0; j < S_; ++j) s += att[i * 44 + j] * vb[(size_t)j * QKVN_];
    o[((size_t)b * RPB_ + i) * D_ + head * HD_ + d] = s;
  }
  for (int idx = tid; idx < (RPB_ - S_) * HD_; idx += 256) {  // zero pad rows
    int i = idx >> 9, d = idx & 511;
    o[((size_t)b * RPB_ + S_ + i) * D_ + head * HD_ + d] = 0.f;
  }
}

// ---------------- scores + min-max + perturbed top-k indicators -------------
__global__ void k_scores(const float* __restrict__ h, const float* __restrict__ sw,
                         const float* __restrict__ sb, const float* __restrict__ noise,
                         float* __restrict__ ind) {
  __shared__ float sc[T_];
  __shared__ int cnt[KSEL_ * T_];
  int b = blockIdx.x, tid = threadIdx.x;
  int wid = tid >> 5, lane = tid & 31;

  for (int t = wid; t < T_; t += 8) {
    const float* hr = h + ((size_t)b * RPB_ + P_ + t) * D_;
    float s = 0.f;
    for (int c = lane; c < D_; c += 32) s += hr[c] * sw[c];
    for (int off = 16; off > 0; off >>= 1) s += __shfl_down(s, off);
    if (lane == 0) sc[t] = s + sb[0];
  }
  __syncthreads();
  if (tid == 0) {
    float lo = sc[0], hi = sc[0];
    for (int t = 1; t < T_; ++t) { lo = fminf(lo, sc[t]); hi = fmaxf(hi, sc[t]); }
    float inv = 1.f / (hi - lo + 1e-5f);
    for (int t = 0; t < T_; ++t) sc[t] = (sc[t] - lo) * inv;
  }
  __syncthreads();
  for (int i = tid; i < KSEL_ * T_; i += 256) cnt[i] = 0;
  __syncthreads();
  for (int smp = tid; smp < NSAMP_; smp += 256) {
    float p[T_];
    const float* nz = noise + ((size_t)b * NSAMP_ + smp) * T_;
    for (int t = 0; t < T_; ++t) p[t] = sc[t] + 0.05f * nz[t];
    unsigned mask = 0u;
    for (int i = 0; i < T_; ++i) {                     // jax top_k tie-break: lower index wins
      int rank = 0;
      for (int j = 0; j < T_; ++j)
        rank += (p[j] > p[i]) || ((p[j] == p[i]) && (j < i));
      if (rank < KSEL_) mask |= (1u << i);
    }
    for (int i = 0; i < T_; ++i)
      if (mask & (1u << i)) {
        int k = __popc(mask & ((1u << i) - 1u));       // sorted-index position
        atomicAdd(&cnt[k * T_ + i], 1);
      }
  }
  __syncthreads();
  for (int i = tid; i < KSEL_ * T_; i += 256)
    ind[(size_t)b * KSEL_ * T_ + i] = cnt[i] * (1.f / NSAMP_);
}

// ---------------- WMMA gather: out[b] = ind[b](16x32) @ x[b](32x150528) -----
__global__ void k_gather(const float* __restrict__ ind, const float* __restrict__ x,
                         float* __restrict__ out) {
  __shared__ float Ai[KSEL_ * 33];
  int b = blockIdx.y, tid = threadIdx.x;
  int wid = tid >> 5, lane = tid & 31;
  for (int i = tid; i < KSEL_ * T_; i += 256)
    Ai[(i >> 5) * 33 + (i & 31)] = ind[(size_t)b * KSEL_ * T_ + i];
  __syncthreads();
  int m = lane & 15, half = lane >> 4, koff = half * 2;
  int col = (blockIdx.x * 8 + wid) * 16 + m;
  const float* xb = x + (size_t)b * T_ * NC_ + col;
  v8f acc;
#pragma unroll
  for (int r = 0; r < 8; ++r) acc[r] = 0.f;
#pragma unroll
  for (int kb = 0; kb < T_; kb += 4) {
    v2f av, bv;
    av.x = Ai[m * 33 + kb + koff];
    av.y = Ai[m * 33 + kb + koff + 1];
    const float* bp = xb + (size_t)(kb + koff) * NC_;
    bv.x = bp[0];
    bv.y = bp[NC_];
    acc = __builtin_amdgcn_wmma_f32_16x16x4_f32(false, av, false, bv,
                                                (short)0, acc, false, false);
  }
  float* ob = out + (size_t)b * KSEL_ * NC_ + col;
#pragma unroll
  for (int r = 0; r < 8; ++r)
    ob[(size_t)(r + half * 8) * NC_] = acc[r];
}

// ---------------- launch ----------------------------------------------------
extern "C" void kernel_launch(void* const* d_in, const int* in_sizes, int n_in,
                              void* d_out, int out_size, void* d_ws, size_t ws_size,
                              hipStream_t stream) {
  const float* x      = (const float*)d_in[0];
  const float* noise  = (const float*)d_in[1];
  const float* prompt = (const float*)d_in[2];
  const float* ln1_g  = (const float*)d_in[3];
  const float* ln1_b  = (const float*)d_in[4];
  const float* qkv_w  = (const float*)d_in[5];
  const float* qkv_b  = (const float*)d_in[6];
  const float* proj_w = (const float*)d_in[7];
  const float* proj_b = (const float*)d_in[8];
  const float* ln2_g  = (const float*)d_in[9];
  const float* ln2_b  = (const float*)d_in[10];
  const float* fc_w   = (const float*)d_in[11];
  const float* fc_b   = (const float*)d_in[12];
  const float* sw     = (const float*)d_in[13];
  const float* sb     = (const float*)d_in[14];
  float* out = (float*)d_out;

  float* ws  = (float*)d_ws;
  float* h   = ws;                 // 1536*1536
  float* z   = ws + 2359296;       // 1536*1536  (reused as attention output o)
  float* qkv = ws + 4718592;       // 1536*4608  (reused as z2)
  float* indw = ws + 11796480;     // 32*16*32
  float* o  = z;                   // alias: z dead after qkv GEMM
  float* z2 = qkv;                 // alias: qkv dead after attention

  k_init_h<<<dim3(B_, 16), 256, 0, stream>>>(h, prompt);
  k_pool<<<B_ * T_, 256, 0, stream>>>(x, h);
  k_ln<<<ROWS_, 256, 0, stream>>>(h, ln1_g, ln1_b, z);
  k_gemm<<<dim3(ROWS_ / 16, QKVN_ / 128), 256, 0, stream>>>(z, qkv_w, qkv_b, nullptr, qkv, QKVN_);
  k_attn<<<dim3(B_, 3), 256, 0, stream>>>(qkv, o);
  k_gemm<<<dim3(ROWS_ / 16, D_ / 128), 256, 0, stream>>>(o, proj_w, proj_b, h, h, D_);
  k_ln<<<ROWS_, 256, 0, stream>>>(h, ln2_g, ln2_b, z2);
  k_gemm<<<dim3(ROWS_ / 16, D_ / 128), 256, 0, stream>>>(z2, fc_w, fc_b, h, h, D_);
  k_scores<<<B_, 256, 0, stream>>>(h, sw, sb, noise, indw);
  k_gather<<<dim3(NC_ / 128, B_), 256, 0, stream>>>(indw, x, out);
}